// CtdetLoss_56504589746323
// MI455X (gfx1250) — compile-verified
//
#include <hip/hip_runtime.h>

typedef __attribute__((ext_vector_type(2))) float v2f;
typedef __attribute__((ext_vector_type(8))) float v8f;

#define HM_NB 2048
#define REG_NB 64
#define NT 256

// ---------------------------------------------------------------------------
// Wave32 reduction of TWO accumulators via one V_WMMA_F32_16X16X4_F32.
// A (16x4, f32): lane m (0-15):  A[m][0]=x, A[m][1]=y
//                lane m+16:      A[m][2]=x, A[m][3]=y       (ISA A layout)
// B (4x16): column 0 selects K={0,2} (x's); column 1 selects K={1,3} (y's).
//   B vgpr0: lanes0-15 -> K=0 row, lanes16-31 -> K=2 row
//   B vgpr1: lanes0-15 -> K=1 row, lanes16-31 -> K=3 row
// => D[m][0] = x(m)+x(m+16) ;  D[m][1] = y(m)+y(m+16)
// D vgpr p: lanes0-15 hold D[p][n], lanes16-31 hold D[p+8][n].
// s = sum_p d[p];  s += shfl_down(s,16):  lane0 = sum(x), lane1 = sum(y).
// Requires EXEC all-ones: call only at full block convergence.
// ---------------------------------------------------------------------------
__device__ __forceinline__ void wave_reduce_pair(float x, float y,
                                                 float& sx, float& sy) {
  const unsigned n = threadIdx.x & 15u;
  v2f a; a.x = x; a.y = y;
  v2f b; b.x = (n == 0u) ? 1.0f : 0.0f;
  b.y = (n == 1u) ? 1.0f : 0.0f;
  v8f c = {};
  c = __builtin_amdgcn_wmma_f32_16x16x4_f32(false, a, false, b,
                                            (short)0, c, false, false);
  float s = c[0] + c[1] + c[2] + c[3] + c[4] + c[5] + c[6] + c[7];
  s += __shfl_down(s, 16);
  sx = __shfl(s, 0);
  sy = __shfl(s, 1);
}

__device__ __forceinline__ float wave_sum(float v) {
#pragma unroll
  for (int o = 16; o > 0; o >>= 1) v += __shfl_down(v, o);
  return v;  // lane 0 holds the sum
}

// Block-level reduce of 3 accumulators -> dst[0..2] (dst[3] = pad).
__device__ __forceinline__ void block_store3(float a, float b, float c,
                                             float* __restrict__ dst) {
  __shared__ float sm[(NT / 32) * 4];
  float ra, rb;
  wave_reduce_pair(a, b, ra, rb);   // v_wmma_f32_16x16x4_f32
  float rc = wave_sum(c);
  const int lane = threadIdx.x & 31;
  const int wave = threadIdx.x >> 5;
  if (lane == 0) {
    sm[wave * 4 + 0] = ra;
    sm[wave * 4 + 1] = rb;
    sm[wave * 4 + 2] = rc;
  }
  __syncthreads();
  if (threadIdx.x == 0) {
    float A = 0.f, B = 0.f, C = 0.f;
#pragma unroll
    for (int w = 0; w < NT / 32; ++w) {
      A += sm[w * 4 + 0];
      B += sm[w * 4 + 1];
      C += sm[w * 4 + 2];
    }
    dst[0] = A; dst[1] = B; dst[2] = C; dst[3] = 0.f;
  }
}

// ---------------------------------------------------------------------------
// Focal loss element: pred x (logit), gt g.
//   pos: (1-p)^2 * g^4 * log(p)       (g==1 -> g^4==1)
//   neg: p^2 * (1-g)^4 * log(1-p)
// Accumulate signed sum (negated at the end), num_pos, and count(p > 0.3).
// ---------------------------------------------------------------------------
__device__ __forceinline__ void hm_elem(float x, float g, float& ls, float& np,
                                        float& fc) {
  const float e = __expf(-x);                       // v_exp_f32
  float p = __builtin_amdgcn_rcpf(1.0f + e);        // v_rcp_f32
  p = fminf(fmaxf(p, 1e-4f), 1.0f - 1e-4f);
  const bool pos = (g == 1.0f);
  const float omp = 1.0f - p;
  const float a  = pos ? omp : p;                   // part1 base
  const float bq = pos ? g : (1.0f - g);            // part2 base
  const float cl = pos ? p : omp;                   // log argument
  const float b2 = bq * bq;
  ls = fmaf((a * a) * (b2 * b2), __logf(cl), ls);   // v_log_f32
  np += pos ? 1.0f : 0.0f;
  fc += (p > 0.3f) ? 1.0f : 0.0f;
}

__global__ __launch_bounds__(NT) void hm_kernel(
    const float4* __restrict__ pred4, const float4* __restrict__ gt4,
    long long n4, const float* __restrict__ pred, const float* __restrict__ gt,
    long long n, float* __restrict__ partials) {
  float ls = 0.f, np = 0.f, fc = 0.f;
  const long long stride = (long long)gridDim.x * NT;
  for (long long i = (long long)blockIdx.x * NT + threadIdx.x; i < n4;
       i += stride) {
    const float4 x = pred4[i];   // global_load_b128
    const float4 g = gt4[i];     // global_load_b128
    hm_elem(x.x, g.x, ls, np, fc);
    hm_elem(x.y, g.y, ls, np, fc);
    hm_elem(x.z, g.z, ls, np, fc);
    hm_elem(x.w, g.w, ls, np, fc);
  }
  // scalar tail (empty for the reference sizes; kept general)
  for (long long i = n4 * 4 + (long long)blockIdx.x * NT + threadIdx.x; i < n;
       i += stride) {
    hm_elem(pred[i], gt[i], ls, np, fc);
  }
  block_store3(ls, np, fc, partials + (long long)blockIdx.x * 4);
}

// ---------------------------------------------------------------------------
// Gather + smooth-L1 for wh and off (shared mask => shared denominator sum).
// pred layout [B,2,H,W]: gathered[b,k,c] = pred[b*2*HW + c*HW + ind[b,k]].
// ---------------------------------------------------------------------------
__global__ __launch_bounds__(NT) void reg_kernel(
    const float* __restrict__ wh_pred, const float* __restrict__ wh_gt,
    const float* __restrict__ off_pred, const float* __restrict__ off_gt,
    const int* __restrict__ mask, const long long* __restrict__ ind, int BK,
    int K, long long HW, float* __restrict__ partials) {
  float whl = 0.f, ofl = 0.f, ms = 0.f;
  const int stride = gridDim.x * NT;
  for (int i = blockIdx.x * NT + threadIdx.x; i < BK; i += stride) {
    const int b = i / K;
    const long long id = ind[i];
    const float m = (float)mask[i];
    const long long base = (long long)b * 2 * HW + id;
#pragma unroll
    for (int c = 0; c < 2; ++c) {
      {
        const float d = (wh_pred[base + (long long)c * HW] - wh_gt[i * 2 + c]) * m;
        const float ad = fabsf(d);
        whl += (ad < 1.0f) ? 0.5f * d * d : ad - 0.5f;
      }
      {
        const float d = (off_pred[base + (long long)c * HW] - off_gt[i * 2 + c]) * m;
        const float ad = fabsf(d);
        ofl += (ad < 1.0f) ? 0.5f * d * d : ad - 0.5f;
      }
    }
    ms += m;
  }
  block_store3(whl, ofl, ms, partials + (long long)blockIdx.x * 4);
}

// ---------------------------------------------------------------------------
// Final deterministic reduction of per-block partials + loss assembly.
// ---------------------------------------------------------------------------
__global__ __launch_bounds__(NT) void finalize_kernel(
    const float* __restrict__ hm_part, const float* __restrict__ reg_part,
    float* __restrict__ out) {
  float ls = 0.f, np = 0.f, fc = 0.f, whl = 0.f, ofl = 0.f, ms = 0.f;
  for (int i = threadIdx.x; i < HM_NB; i += NT) {
    ls += hm_part[i * 4 + 0];
    np += hm_part[i * 4 + 1];
    fc += hm_part[i * 4 + 2];
  }
  for (int i = threadIdx.x; i < REG_NB; i += NT) {
    whl += reg_part[i * 4 + 0];
    ofl += reg_part[i * 4 + 1];
    ms  += reg_part[i * 4 + 2];
  }
  __shared__ float red[NT];
  auto bsum = [&](float v) -> float {
    red[threadIdx.x] = v;
    __syncthreads();
    for (int s = NT / 2; s > 0; s >>= 1) {
      if ((int)threadIdx.x < s) red[threadIdx.x] += red[threadIdx.x + s];
      __syncthreads();
    }
    const float r = red[0];
    __syncthreads();
    return r;
  };
  ls = bsum(ls); np = bsum(np); fc = bsum(fc);
  whl = bsum(whl); ofl = bsum(ofl); ms = bsum(ms);
  if (threadIdx.x == 0) {
    const float denom = (np == 0.0f) ? fmaxf(fc, 1.0f) : np;
    const float hm_loss = -ls / denom;
    const float rden = 2.0f * ms + 1e-4f;  // mask broadcast over C=2
    const float wh_loss = whl / rden;
    const float off_loss = ofl / rden;
    out[0] = hm_loss;
    out[1] = wh_loss;
    out[2] = off_loss;
    out[3] = 1.0f * hm_loss + 0.1f * wh_loss + 1.0f * off_loss;
  }
}

extern "C" void kernel_launch(void* const* d_in, const int* in_sizes, int n_in,
                              void* d_out, int out_size, void* d_ws,
                              size_t ws_size, hipStream_t stream) {
  (void)n_in; (void)out_size; (void)ws_size;
  const float* hm_pred  = (const float*)d_in[0];
  const float* hm_gt    = (const float*)d_in[1];
  const float* wh_pred  = (const float*)d_in[2];
  const float* wh_gt    = (const float*)d_in[3];
  const float* off_pred = (const float*)d_in[4];
  const float* off_gt   = (const float*)d_in[5];
  const int* mask       = (const int*)d_in[6];
  const long long* ind  = (const long long*)d_in[7];
  float* out = (float*)d_out;
  float* ws  = (float*)d_ws;

  const long long N = in_sizes[0];        // B*C*H*W (41.9M)
  const int BK = in_sizes[6];             // B*K
  const int K = 128;                      // reference: max objects
  const int B = BK / K;
  const long long HW = (long long)in_sizes[2] / (2LL * B);

  float* hm_part  = ws;                   // HM_NB * 4 floats
  float* reg_part = ws + HM_NB * 4;       // REG_NB * 4 floats

  hm_kernel<<<HM_NB, NT, 0, stream>>>((const float4*)hm_pred,
                                      (const float4*)hm_gt, N >> 2, hm_pred,
                                      hm_gt, N, hm_part);
  reg_kernel<<<REG_NB, NT, 0, stream>>>(wh_pred, wh_gt, off_pred, off_gt, mask,
                                        ind, BK, K, HW, reg_part);
  finalize_kernel<<<1, NT, 0, stream>>>(hm_part, reg_part, out);
}